// Codebook_8976481649073
// MI455X (gfx1250) — compile-verified
//
#include <hip/hip_runtime.h>

typedef __attribute__((ext_vector_type(2))) float v2f;
typedef __attribute__((ext_vector_type(8))) float v8f;

#define KD      64
#define KPAD    68          // LDS row pad: 68*4B = 272B -> stride 4 banks, no 16-way conflict
#define NCODES  2048
#define NROWS   32768
#define HALFR   16384
#define EPSF    1e-8f

__device__ __forceinline__ unsigned orderKey(float f) {
    unsigned u = __float_as_uint(f);
    return (u & 0x80000000u) ? ~u : (u | 0x80000000u);
}
__device__ __forceinline__ float unorderKey(unsigned u) {
    unsigned b = (u & 0x80000000u) ? (u & 0x7FFFFFFFu) : ~u;
    return __uint_as_float(b);
}

// ---------------- prep: emb row norms + zero scalar accumulators ----------------
__global__ void vq_prep(const float* __restrict__ emb, float* __restrict__ embnorm,
                        float* __restrict__ acc3) {
    int n = blockIdx.x * 256 + threadIdx.x;
    if (n < NCODES) {
        const float* e = emb + n * KD;
        float s = 0.f;
        #pragma unroll
        for (int k = 0; k < KD; ++k) s += e[k] * e[k];
        embnorm[n] = s;
    }
    if (blockIdx.x == 0 && threadIdx.x < 3) acc3[threadIdx.x] = 0.f;
}

// ---------------- fused main kernel ----------------
// block: 1024 threads (32 waves). Handles rows [m0,m0+16) (tile0/p1) and
// [m0+16384, m0+16400) (tile1/p2). Each wave computes 4 n-tiles of 16 cols.
__global__ __launch_bounds__(1024)
void vq_main(const float* __restrict__ z, const float* __restrict__ emb,
             const float* __restrict__ embnorm,
             float* __restrict__ zq_out, float* __restrict__ dp_out,
             float* __restrict__ acc3) {
    __shared__ float lds_z[2][16][KPAD];
    __shared__ unsigned long long minpack[2][16];
    __shared__ float rowsum[2][16];
    __shared__ float znorm[2][16];
    __shared__ float blk_acc[3];     // kl, ent, mse

    const int tid  = threadIdx.x;
    const int wave = tid >> 5;
    const int lane = tid & 31;
    const int lm   = lane & 15;      // N (or M) position within tile
    const int hs   = lane >> 4;      // lane half select
    const int m0   = blockIdx.x * 16;

    // ---- stage z rows into LDS (z layout: z[b][k][h][w]; row m -> b=m>>12,h=(m>>6)&63,w=m&63)
    for (int i = tid; i < 2 * 16 * KD; i += 1024) {
        int k = i >> 5, r = i & 31;
        int tile = r >> 4, rr = r & 15;
        int mg = m0 + rr + tile * HALFR;
        int b = mg >> 12, h = (mg >> 6) & 63, w = mg & 63;
        lds_z[tile][rr][k] = z[(((size_t)b * 64 + k) * 64 + h) * 64 + w];
    }
    if (tid < 32) {
        minpack[tid >> 4][tid & 15] = ~0ull;
        rowsum[tid >> 4][tid & 15]  = 0.f;
        znorm[tid >> 4][tid & 15]   = 0.f;
    }
    if (tid < 3) blk_acc[tid] = 0.f;
    __syncthreads();

    // ---- z row norms
    for (int i = tid; i < 2 * 16 * 16; i += 1024) {
        int r = i & 31, kk = i >> 5;
        int tile = r >> 4, rr = r & 15;
        const float* zr = &lds_z[tile][rr][kk * 4];
        atomicAdd(&znorm[tile][rr], zr[0]*zr[0] + zr[1]*zr[1] + zr[2]*zr[2] + zr[3]*zr[3]);
    }
    __syncthreads();

    // ---- GEMM: dist = |z|^2 + |e|^2 - 2 z.e  via V_WMMA_F32_16X16X4_F32, K=64
    v8f acc0[4], acc1[4];
    #pragma unroll
    for (int t = 0; t < 4; ++t) {
        const int n0 = (wave * 4 + t) * 16;
        const float* bp = emb + (size_t)(n0 + lm) * KD + 2 * hs;  // B = emb^T fragment base
        v8f c0 = {}, c1 = {};
        #pragma unroll
        for (int kk = 0; kk < 16; ++kk) {
            const int kb = kk * 4 + 2 * hs;
            v2f a0; a0.x = lds_z[0][lm][kb]; a0.y = lds_z[0][lm][kb + 1];
            v2f a1; a1.x = lds_z[1][lm][kb]; a1.y = lds_z[1][lm][kb + 1];
            v2f bf; bf.x = bp[kk * 4];       bf.y = bp[kk * 4 + 1];
            c0 = __builtin_amdgcn_wmma_f32_16x16x4_f32(false, a0, false, bf, (short)0, c0, false, false);
            c1 = __builtin_amdgcn_wmma_f32_16x16x4_f32(false, a1, false, bf, (short)0, c1, false, false);
        }
        acc0[t] = c0; acc1[t] = c1;
    }

    // convert accumulators to distances (C/D layout: M = v + 8*hs, N = lm)
    #pragma unroll
    for (int t = 0; t < 4; ++t) {
        const int n0 = (wave * 4 + t) * 16;
        const float en = embnorm[n0 + lm];
        #pragma unroll
        for (int v = 0; v < 8; ++v) {
            const int M = v + 8 * hs;
            acc0[t][v] = znorm[0][M] + en - 2.0f * acc0[t][v];
            acc1[t][v] = znorm[1][M] + en - 2.0f * acc1[t][v];
        }
    }

    // ---- per-row min + argmin (packed u64, ties -> lowest index, matches argmin)
    #pragma unroll
    for (int v = 0; v < 8; ++v) {
        const int M = v + 8 * hs;
        float bd0 = acc0[0][v], bd1 = acc1[0][v];
        unsigned bn0 = wave * 64 + lm, bn1 = bn0;
        #pragma unroll
        for (int t = 1; t < 4; ++t) {
            unsigned n = wave * 64 + t * 16 + lm;
            float d0 = acc0[t][v]; if (d0 < bd0) { bd0 = d0; bn0 = n; }
            float d1 = acc1[t][v]; if (d1 < bd1) { bd1 = d1; bn1 = n; }
        }
        unsigned long long k0 = ((unsigned long long)orderKey(bd0) << 32) | bn0;
        unsigned long long k1 = ((unsigned long long)orderKey(bd1) << 32) | bn1;
        #pragma unroll
        for (int m = 8; m >= 1; m >>= 1) {
            unsigned long long o0 = __shfl_xor(k0, m); if (o0 < k0) k0 = o0;
            unsigned long long o1 = __shfl_xor(k1, m); if (o1 < k1) k1 = o1;
        }
        if (lm == 0) {
            atomicMin(&minpack[0][M], k0);
            atomicMin(&minpack[1][M], k1);
        }
    }
    __syncthreads();

    // ---- softmax: e = exp(min - dist); accumulate row sums (overwrite acc with e)
    #pragma unroll
    for (int v = 0; v < 8; ++v) {
        const int M = v + 8 * hs;
        const float rm0 = unorderKey((unsigned)(minpack[0][M] >> 32));
        const float rm1 = unorderKey((unsigned)(minpack[1][M] >> 32));
        float s0 = 0.f, s1 = 0.f;
        #pragma unroll
        for (int t = 0; t < 4; ++t) {
            float e0 = __expf(rm0 - acc0[t][v]);
            float e1 = __expf(rm1 - acc1[t][v]);
            acc0[t][v] = e0; acc1[t][v] = e1;
            s0 += e0; s1 += e1;
        }
        #pragma unroll
        for (int m = 8; m >= 1; m >>= 1) { s0 += __shfl_xor(s0, m); s1 += __shfl_xor(s1, m); }
        if (lm == 0) { atomicAdd(&rowsum[0][M], s0); atomicAdd(&rowsum[1][M], s1); }
    }
    __syncthreads();

    // ---- normalize, stream probs out (NT), accumulate JSD/entropy terms
    float kl_part = 0.f, ent_part = 0.f;
    #pragma unroll
    for (int v = 0; v < 8; ++v) {
        const int M = v + 8 * hs;
        const float inv0 = 1.0f / rowsum[0][M];
        const float inv1 = 1.0f / rowsum[1][M];
        float* r0 = dp_out + (size_t)(m0 + M) * NCODES;
        float* r1 = dp_out + (size_t)(m0 + HALFR + M) * NCODES;
        #pragma unroll
        for (int t = 0; t < 4; ++t) {
            const int n = wave * 64 + t * 16 + lm;
            float p1 = acc0[t][v] * inv0;
            float p2 = acc1[t][v] * inv1;
            __builtin_nontemporal_store(p1, r0 + n);
            __builtin_nontemporal_store(p2, r1 + n);
            float lmid = __logf(0.5f * (p1 + p2) + EPSF);
            float l1 = __logf(p1 + EPSF);
            float l2 = __logf(p2 + EPSF);
            kl_part  += p1 * (l1 - lmid) + p2 * (l2 - lmid);
            ent_part += -p1 * l1 - p2 * l2;
        }
    }

    // ---- z_q gather, z_q_out (straight-through forward == z_q), mse partials
    float mse_part = 0.f;
    {
        const int r = tid & 31, dch = tid >> 5;        // 32 rows x 32 channel-threads (x2 channels)
        const int tile = r >> 4, rr = r & 15;
        const int mg = m0 + rr + tile * HALFR;
        const int b = mg >> 12, h = (mg >> 6) & 63, w = mg & 63;
        const unsigned idx = (unsigned)(minpack[tile][rr] & 0xFFFFFFFFull);
        const float* er = emb + (size_t)idx * KD;
        #pragma unroll
        for (int dd = 0; dd < 2; ++dd) {
            const int d = dch + dd * 32;
            float q = er[d];
            float diff = q - lds_z[tile][rr][d];
            mse_part += diff * diff;
            __builtin_nontemporal_store(q, zq_out + (((size_t)b * 64 + d) * 64 + h) * 64 + w);
        }
    }

    // ---- block reduction of scalars -> global atomics
    #pragma unroll
    for (int m = 16; m >= 1; m >>= 1) {
        kl_part  += __shfl_xor(kl_part, m);
        ent_part += __shfl_xor(ent_part, m);
        mse_part += __shfl_xor(mse_part, m);
    }
    if (lane == 0) {
        atomicAdd(&blk_acc[0], kl_part);
        atomicAdd(&blk_acc[1], ent_part);
        atomicAdd(&blk_acc[2], mse_part);
    }
    __syncthreads();
    if (tid == 0) {
        atomicAdd(&acc3[0], blk_acc[0]);
        atomicAdd(&acc3[1], blk_acc[1]);
        atomicAdd(&acc3[2], blk_acc[2]);
    }
}

// ---------------- finalize scalars ----------------
__global__ void vq_fin(const float* __restrict__ acc3, float* __restrict__ scal) {
    if (threadIdx.x == 0) {
        scal[0] = 1.25f * acc3[2] / (float)(NROWS * KD);   // q_loss = (1+beta)*mse
        scal[1] = 0.5f * acc3[0] / (float)HALFR;           // jsd
        scal[2] = 0.5f * acc3[1] / (float)HALFR;           // entropy
    }
}

extern "C" void kernel_launch(void* const* d_in, const int* in_sizes, int n_in,
                              void* d_out, int out_size, void* d_ws, size_t ws_size,
                              hipStream_t stream) {
    const float* z   = (const float*)d_in[0];
    const float* emb = (const float*)d_in[1];
    float* out = (float*)d_out;

    float* embnorm = (float*)d_ws;             // 2048 floats
    float* acc3    = embnorm + NCODES;         // 3 floats

    float* zq   = out;                         // 8*64*64*64 = 2097152
    float* scal = out + 2097152;               // q_loss, jsd, entropy
    float* dp   = out + 2097155;               // 32768*2048

    vq_prep<<<8, 256, 0, stream>>>(emb, embnorm, acc3);
    vq_main<<<NROWS / 32, 1024, 0, stream>>>(z, emb, embnorm, zq, dp, acc3);
    vq_fin<<<1, 32, 0, stream>>>(acc3, scal);
}